// Net_40063454937540
// MI455X (gfx1250) — compile-verified
//
#include <hip/hip_runtime.h>
#include <hip/hip_bf16.h>
#include <math.h>

typedef float v2f __attribute__((ext_vector_type(2)));
typedef float v8f __attribute__((ext_vector_type(8)));

#define F_IN 512
#define HID  8
#define NC   64

__device__ __forceinline__ float lrelu02(float a) { return a > 0.0f ? a : 0.2f * a; }
__device__ __forceinline__ float elu1(float a)    { return a > 0.0f ? a : (expf(a) - 1.0f); }

// ---------------- utility ----------------
__global__ void zero_f32(float* __restrict__ p, long long n) {
  long long i = (long long)blockIdx.x * blockDim.x + threadIdx.x;
  if (i < n) p[i] = 0.0f;
}

// int64 edge index -> int32 row/col, degree of source nodes
__global__ void prep_edges(const long long* __restrict__ ei, int E,
                           int* __restrict__ row32, int* __restrict__ col32,
                           float* __restrict__ deg) {
  int e = blockIdx.x * blockDim.x + threadIdx.x;
  if (e >= E) return;
  int r = (int)ei[e];
  int c = (int)ei[(long long)E + e];
  row32[e] = r;
  col32[e] = c;
  atomicAdd(&deg[r], 1.0f);
}

__global__ void dinv_k(const float* __restrict__ deg, float* __restrict__ dinv, int N) {
  int i = blockIdx.x * blockDim.x + threadIdx.x;
  if (i >= N) return;
  float d = deg[i];
  dinv[i] = d > 0.0f ? (1.0f / sqrtf(d)) : 0.0f;
}

__global__ void norm_k(const int* __restrict__ row32, const int* __restrict__ col32,
                       const float* __restrict__ dinv, float* __restrict__ nrm, int E) {
  int e = blockIdx.x * blockDim.x + threadIdx.x;
  if (e >= E) return;
  nrm[e] = dinv[row32[e]] * dinv[col32[e]];
}

// Collapsed edge-MLP constants: c = lrelu_slopes(mWa) @ mWb  (valid since norm >= 0)
__global__ void cmlp_k(const float* __restrict__ mW1a, const float* __restrict__ mW1b,
                       const float* __restrict__ mW2a, const float* __restrict__ mW2b,
                       float* __restrict__ c1, float* __restrict__ c2) {
  int j = threadIdx.x;
  if (j < HID) {
    float s = 0.0f;
    for (int i = 0; i < HID; ++i) s += lrelu02(mW1a[i]) * mW1b[i * HID + j];
    c1[j] = s;
  }
  if (j < NC) {
    float s = 0.0f;
    for (int i = 0; i < NC; ++i) s += lrelu02(mW2a[i]) * mW2b[i * NC + j];
    c2[j] = s;
  }
}

// Pre-pack B operand in per-lane WMMA layout, zero padding baked in.
// Bp[(s*ntiles + nt)*32 + lane] = { W[kr][n], W[kr+1][n] },
//   kr = 4*s + (lane>>4)*2,  n = nt*16 + (lane&15),  OOB columns -> 0.
__global__ void packB_k(const float* __restrict__ W, int K, int C, int ntiles,
                        v2f* __restrict__ Bp) {
  int t = blockIdx.x * blockDim.x + threadIdx.x;
  int total = (K / 4) * ntiles * 32;
  if (t >= total) return;
  int lane = t & 31;
  int nt   = (t >> 5) % ntiles;
  int s    = t / (32 * ntiles);
  int kr = 4 * s + (lane >> 4) * 2;
  int n  = nt * 16 + (lane & 15);
  v2f v;
  v.x = (n < C) ? W[kr * C + n]       : 0.0f;
  v.y = (n < C) ? W[(kr + 1) * C + n] : 0.0f;
  Bp[t] = v;
}

// ---------------- GEMM1: h1 = x @ W1 + b1   (N x 512) @ (512 x 8), WMMA f32 16x16x4 ----------------
__global__ __launch_bounds__(256) void gemm1_wmma(const float* __restrict__ x,
                                                  const v2f* __restrict__ Bp1,
                                                  const float* __restrict__ b1,
                                                  float* __restrict__ h1, int N) {
  const int lane = threadIdx.x & 31;
  const int wave = threadIdx.x >> 5;
  const int tile = blockIdx.x * 8 + wave;     // 16-row M tile per wave
  const int m0 = tile * 16;
  if (m0 >= N) return;

  const int half = lane >> 4;                 // 0: K{0,1}, 1: K{2,3}
  const int lm   = lane & 15;                 // A: M row; B/C/D: N col

  int row = m0 + lm; if (row >= N) row = N - 1;          // clamp OOB rows (not stored)
  const float* __restrict__ xrow = x + (size_t)row * F_IN + half * 2;

  v8f acc = {0.f,0.f,0.f,0.f,0.f,0.f,0.f,0.f};

  for (int k0 = 0; k0 < F_IN; k0 += 64) {
    __builtin_prefetch(xrow + k0 + 64, 0, 3); // pull next chunk into WGP caches
#pragma unroll
    for (int kk = 0; kk < 64; kk += 4) {
      const int k = k0 + kk;
      v2f a = *(const v2f*)(xrow + k);        // contiguous pair -> global_load_b64
      v2f b = Bp1[(k >> 2) * 32 + lane];      // pre-packed, unconditional, coalesced
      acc = __builtin_amdgcn_wmma_f32_16x16x4_f32(false, a, false, b,
                                                  (short)0, acc, false, false);
    }
  }

  const int n = lm;                           // output column (only n<8 valid)
  if (n < HID) {
    const float bias = b1[n];
#pragma unroll
    for (int r = 0; r < 8; ++r) {
      int m = m0 + r + half * 8;              // C/D layout: M = r + 8*(lane>=16)
      if (m < N) h1[(size_t)m * HID + n] = acc[r] + bias;
    }
  }
}

// ---------------- GEMM2: h2 = elu(agg1) @ W2 + b2   (N x 8) @ (8 x 64), WMMA f32 ----------------
__global__ __launch_bounds__(256) void gemm2_wmma(const float* __restrict__ agg1,
                                                  const v2f* __restrict__ Bp2,
                                                  const float* __restrict__ b2,
                                                  float* __restrict__ h2, int N) {
  const int lane = threadIdx.x & 31;
  const int wave = threadIdx.x >> 5;
  const int tile = blockIdx.x * 8 + wave;
  const int m0 = tile * 16;
  if (m0 >= N) return;

  const int half = lane >> 4;
  const int lm   = lane & 15;

  int row = m0 + lm; if (row >= N) row = N - 1;
  const float* __restrict__ arow = agg1 + (size_t)row * HID + half * 2;

  v8f acc0 = {0.f,0.f,0.f,0.f,0.f,0.f,0.f,0.f};
  v8f acc1 = acc0, acc2 = acc0, acc3 = acc0;

#pragma unroll
  for (int k = 0; k < HID; k += 4) {
    v2f a;
    a.x = elu1(arow[k + 0]);                  // fused ELU on layer-1 output
    a.y = elu1(arow[k + 1]);
    const int s = k >> 2;
#pragma unroll
    for (int nt = 0; nt < 4; ++nt) {
      v2f b = Bp2[(s * 4 + nt) * 32 + lane];
      v8f c = (nt == 0) ? acc0 : (nt == 1) ? acc1 : (nt == 2) ? acc2 : acc3;
      c = __builtin_amdgcn_wmma_f32_16x16x4_f32(false, a, false, b,
                                                (short)0, c, false, false);
      if (nt == 0) acc0 = c; else if (nt == 1) acc1 = c;
      else if (nt == 2) acc2 = c; else acc3 = c;
    }
  }

#pragma unroll
  for (int nt = 0; nt < 4; ++nt) {
    const int n = nt * 16 + lm;
    const float bias = b2[n];
    const v8f c = (nt == 0) ? acc0 : (nt == 1) ? acc1 : (nt == 2) ? acc2 : acc3;
#pragma unroll
    for (int r = 0; r < 8; ++r) {
      int m = m0 + r + half * 8;
      if (m < N) h2[(size_t)m * NC + n] = c[r] + bias;
    }
  }
}

// ---------------- edge passes: msg = (norm*c + mb) * h[row], scatter-add to [col] ----------------
__global__ void edge1_k(const int* __restrict__ row32, const int* __restrict__ col32,
                        const float* __restrict__ nrm, const float* __restrict__ h1,
                        const float* __restrict__ c1, const float* __restrict__ mb1,
                        float* __restrict__ agg1, int E) {
  long long t = (long long)blockIdx.x * blockDim.x + threadIdx.x;
  if (t >= (long long)E * HID) return;
  int e = (int)(t >> 3);
  int c = (int)(t & (HID - 1));
  float w = nrm[e] * c1[c] + mb1[c];
  int r = row32[e], d = col32[e];
  atomicAdd(&agg1[(long long)d * HID + c], w * h1[(long long)r * HID + c]);
}

__global__ void edge2_k(const int* __restrict__ row32, const int* __restrict__ col32,
                        const float* __restrict__ nrm, const float* __restrict__ h2,
                        const float* __restrict__ c2, const float* __restrict__ mb2,
                        float* __restrict__ out, int E) {
  long long t = (long long)blockIdx.x * blockDim.x + threadIdx.x;
  if (t >= (long long)E * NC) return;
  int e = (int)(t >> 6);
  int c = (int)(t & (NC - 1));
  float w = nrm[e] * c2[c] + mb2[c];
  int r = row32[e], d = col32[e];
  atomicAdd(&out[(long long)d * NC + c], w * h2[(long long)r * NC + c]);
}

// ---------------- log_softmax over 64 channels, one wave32 per node, in place ----------------
__global__ void lsm_k(float* __restrict__ out, int N) {
  int node = (int)(((long long)blockIdx.x * blockDim.x + threadIdx.x) >> 5);
  int lane = threadIdx.x & 31;
  if (node >= N) return;
  float* p = out + (size_t)node * NC;
  float v0 = p[lane], v1 = p[lane + 32];
  float m = fmaxf(v0, v1);
#pragma unroll
  for (int s = 16; s > 0; s >>= 1) m = fmaxf(m, __shfl_xor(m, s, 32));
  float sum = expf(v0 - m) + expf(v1 - m);
#pragma unroll
  for (int s = 16; s > 0; s >>= 1) sum += __shfl_xor(sum, s, 32);
  float l = m + logf(sum);
  p[lane] = v0 - l;
  p[lane + 32] = v1 - l;
}

// ---------------- host ----------------
extern "C" void kernel_launch(void* const* d_in, const int* in_sizes, int n_in,
                              void* d_out, int out_size, void* d_ws, size_t ws_size,
                              hipStream_t stream) {
  const float*     x    = (const float*)d_in[0];
  const long long* ei   = (const long long*)d_in[1];
  const float*     W1   = (const float*)d_in[2];
  const float*     b1   = (const float*)d_in[3];
  const float*     mW1a = (const float*)d_in[4];
  const float*     mW1b = (const float*)d_in[5];
  const float*     mb1  = (const float*)d_in[6];
  const float*     W2   = (const float*)d_in[7];
  const float*     b2   = (const float*)d_in[8];
  const float*     mW2a = (const float*)d_in[9];
  const float*     mW2b = (const float*)d_in[10];
  const float*     mb2  = (const float*)d_in[11];
  float* out = (float*)d_out;

  const int N = in_sizes[0] / F_IN;
  const int E = in_sizes[1] / 2;

  // workspace carve-up (256B aligned)
  char* ws = (char*)d_ws;
  size_t off = 0;
  auto carve = [&](size_t bytes) -> void* {
    void* p = ws + off;
    off += (bytes + 255) & ~(size_t)255;
    return p;
  };
  int*   row32 = (int*)  carve((size_t)E * 4);
  int*   col32 = (int*)  carve((size_t)E * 4);
  float* deg   = (float*)carve((size_t)N * 4);
  float* dinv  = (float*)carve((size_t)N * 4);
  float* nrm   = (float*)carve((size_t)E * 4);
  float* h1    = (float*)carve((size_t)N * HID * 4);
  float* agg1  = (float*)carve((size_t)N * HID * 4);
  float* h2    = (float*)carve((size_t)N * NC * 4);
  float* c1    = (float*)carve(HID * 4);
  float* c2    = (float*)carve(NC * 4);
  v2f*   Bp1   = (v2f*)  carve((size_t)(F_IN / 4) * 1 * 32 * 8);  // 32 KB
  v2f*   Bp2   = (v2f*)  carve((size_t)(HID / 4) * 4 * 32 * 8);   //  2 KB
  (void)ws_size; (void)n_in; (void)out_size;

  const int B = 256;
  auto blocks = [](long long n, int b) { return (unsigned)((n + b - 1) / b); };

  // zero accumulators
  zero_f32<<<blocks(N, B), B, 0, stream>>>(deg, N);
  zero_f32<<<blocks((long long)N * HID, B), B, 0, stream>>>(agg1, (long long)N * HID);
  zero_f32<<<blocks((long long)N * NC, B), B, 0, stream>>>(out, (long long)N * NC);

  // graph preprocessing + operand packing
  prep_edges<<<blocks(E, B), B, 0, stream>>>(ei, E, row32, col32, deg);
  dinv_k<<<blocks(N, B), B, 0, stream>>>(deg, dinv, N);
  norm_k<<<blocks(E, B), B, 0, stream>>>(row32, col32, dinv, nrm, E);
  cmlp_k<<<1, 64, 0, stream>>>(mW1a, mW1b, mW2a, mW2b, c1, c2);
  packB_k<<<blocks((F_IN / 4) * 1 * 32, B), B, 0, stream>>>(W1, F_IN, HID, 1, Bp1);
  packB_k<<<blocks((HID / 4) * 4 * 32, B), B, 0, stream>>>(W2, HID, NC, 4, Bp2);

  // layer 1
  const unsigned gtiles = (unsigned)(((N + 15) / 16 + 7) / 8);
  gemm1_wmma<<<gtiles, 256, 0, stream>>>(x, Bp1, b1, h1, N);
  edge1_k<<<blocks((long long)E * HID, B), B, 0, stream>>>(row32, col32, nrm, h1, c1, mb1, agg1, E);

  // layer 2 (ELU fused into GEMM2 A-operand)
  gemm2_wmma<<<gtiles, 256, 0, stream>>>(agg1, Bp2, b2, h2, N);
  edge2_k<<<blocks((long long)E * NC, B), B, 0, stream>>>(row32, col32, nrm, h2, c2, mb2, out, E);

  // log_softmax in place
  lsm_k<<<blocks((long long)N * 32, B), B, 0, stream>>>(out, N);
}